// TableBatchedTTEmbeddingBag_84576495993304
// MI455X (gfx1250) — compile-verified
//
#include <hip/hip_runtime.h>

// ---------------------------------------------------------------------------
// TT embedding bag, two-phase WMMA formulation for gfx1250 (MI455X).
//
// Phase 1: w[t][i0g][i1](16x128) = core0[t][4*i0g:4*i0g+4] (as 16x32, dense)
//                                @ core1[t][i1]            (as 32x128, dense)
//          -> 10,000 dense 16x128 GEMMs, v_wmma_f32_16x16x4_f32, 100% tile util.
//          Result table: 10,000 * 2048 floats = 81,920,000 B in d_ws
//          (fits the 192 MB global L2 -> phase-2 gathers never hit HBM).
//
// Phase 2: per bag, pooling fused into K:
//          out_bag(16x4) = A(16x512) @ B(512x4)
//          A cols (lookup j, s): t_j[(a,b), s] = w-slice rows,
//          B rows (j, s):        c2_j[s, c]   (cols 4..15 zero-padded).
//          128 K-steps of v_wmma_f32_16x16x4_f32 per bag, accumulator held
//          in 8 VGPRs across all 16 lookups.
// ---------------------------------------------------------------------------

typedef float v2f __attribute__((ext_vector_type(2)));
typedef float v8f __attribute__((ext_vector_type(8)));

#define NTAB   4
#define NP     100
#define I0G    25            // 100 i0 values in groups of 4 -> dense M=16
#define WSLICE 2048          // 16x128 floats per (t, i0g, i1)

// ---------------------------------------------------------------------------
// Phase 1: dense 16x128 = 16x32 @ 32x128 per block, 8 waves x one 16x16 tile.
// ---------------------------------------------------------------------------
__global__ __launch_bounds__(256) void tt_phase1_kernel(
    const float* __restrict__ core0,   // [4][100][128]
    const float* __restrict__ core1,   // [4][100][4096]
    float* __restrict__ w)             // [4][25][100][16][128]
{
    __shared__ float lA[16 * 32];      // 2 KB
    __shared__ float lB[32 * 128];     // 16 KB

    const int bid = blockIdx.x;                 // [0, 10000)
    const int t   = bid / (I0G * NP);
    const int g   = (bid / NP) % I0G;
    const int i1  = bid % NP;

    const float* A = core0 + ((size_t)t * NP + (size_t)g * 4) * 128; // 16x32
    const float* B = core1 + ((size_t)t * NP + i1) * 4096;           // 32x128

    const int tid = threadIdx.x;

    // cooperative stage into LDS
    reinterpret_cast<float2*>(lA)[tid] = reinterpret_cast<const float2*>(A)[tid];
#pragma unroll
    for (int i = 0; i < 4; ++i)
        reinterpret_cast<float4*>(lB)[i * 256 + tid] =
            reinterpret_cast<const float4*>(B)[i * 256 + tid];
    __syncthreads();

    const int wave = tid >> 5;
    const int lane = tid & 31;
    const int hi   = lane >> 4;        // 0: lanes 0-15 (K0/K1), 1: lanes 16-31 (K2/K3)
    const int lrow = lane & 15;
    const int col  = wave * 16 + lrow; // this wave's 16-wide N tile

    v8f acc = {};
#pragma unroll
    for (int kk = 0; kk < 8; ++kk) {   // K = 32 in steps of 4
        const int r0 = kk * 4;
        // A frag: lane<16 -> (K0,K1) of row lrow; lane>=16 -> (K2,K3)
        v2f a = *reinterpret_cast<const v2f*>(&lA[lrow * 32 + r0 + 2 * hi]);
        // B frag: mirrored K split across lane halves / VGPRs
        v2f b;
        b.x = lB[(r0 + 2 * hi    ) * 128 + col];
        b.y = lB[(r0 + 2 * hi + 1) * 128 + col];
        acc = __builtin_amdgcn_wmma_f32_16x16x4_f32(
            /*neg_a=*/false, a, /*neg_b=*/false, b,
            /*c_mod=*/(short)0, acc, /*reuse_a=*/false, /*reuse_b=*/false);
    }

    // C layout: VGPR v, lanes 0-15 -> M=v ; lanes 16-31 -> M=v+8
    float* wout = w + (((size_t)t * I0G + g) * NP + i1) * (size_t)WSLICE;
#pragma unroll
    for (int v = 0; v < 8; ++v)
        wout[(v + 8 * hi) * 128 + col] = acc[v];
}

// ---------------------------------------------------------------------------
// Phase 2: one wave per bag; 16 lookups fused into K=512; acc in 8 VGPRs.
// ---------------------------------------------------------------------------
__global__ __launch_bounds__(256) void tt_phase2_kernel(
    const int*   __restrict__ indices,
    const int*   __restrict__ offsets,
    const float* __restrict__ core2,   // [4][100][128]  ([s][c] per row)
    const float* __restrict__ w,       // phase-1 table (L2-resident)
    float* __restrict__ out)           // [32768][64]
{
    __shared__ float lc2[8][128];      // per-wave c2 slice strip (4 KB)

    const int wave = threadIdx.x >> 5;
    const int lane = threadIdx.x & 31;
    const int hi   = lane >> 4;
    const int lrow = lane & 15;
    const int bag  = blockIdx.x * 8 + wave;
    const int t    = bag >> 13;                  // bags_per_table = 8192
    const int start = offsets[bag];
    const int end   = offsets[bag + 1];
    const bool cval = (lrow < 4);                // only N cols 0..3 are real

    v8f acc = {};
    for (int j = start; j < end; ++j) {          // wave-uniform loop
        const int idx = indices[j];
        const int i0  = idx / (NP * NP);
        const int i1  = (idx / NP) % NP;
        const int i2  = idx % NP;

        // stage c2 slice (32x4 f32 = 512 B) into this wave's LDS strip
        reinterpret_cast<float4*>(lc2[wave])[lane] =
            reinterpret_cast<const float4*>(core2 + ((size_t)t * NP + i2) * 128)[lane];
        asm volatile("s_wait_dscnt 0x0" ::: "memory");  // wave-local producer->consumer

        const float* ws = w + (((size_t)t * I0G + (i0 >> 2)) * NP + i1) * (size_t)WSLICE;
        const int n0    = i0 & 3;
        // A row (a,b) = lrow -> w-slice row (n0*4 + a), col b*32 + s
        const int abase = (n0 * 4 + (lrow >> 2)) * 128 + (lrow & 3) * 32 + 2 * hi;

        if (j + 1 < end) {                        // uniform branch (EXEC stays ~0)
            const int nidx = indices[j + 1];
            const int ni0  = nidx / (NP * NP);
            const int ni1  = (nidx / NP) % NP;
            __builtin_prefetch(                    // global_prefetch_b8: warm next A-slice
                w + (((size_t)t * I0G + (ni0 >> 2)) * NP + ni1) * (size_t)WSLICE, 0, 0);
        }

#pragma unroll
        for (int kk = 0; kk < 8; ++kk) {          // K = 32 per lookup, steps of 4
            const int s0 = kk * 4;
            // A gather: one global b64 per lane; each w byte read exactly once
            v2f a = *reinterpret_cast<const v2f*>(ws + abase + s0);
            // B frag from LDS, zero-padded cols 4..15 via cndmask (EXEC all-1s)
            const float b0 = lc2[wave][(s0 + 2 * hi    ) * 4 + (lrow & 3)];
            const float b1 = lc2[wave][(s0 + 2 * hi + 1) * 4 + (lrow & 3)];
            v2f b;
            b.x = cval ? b0 : 0.0f;
            b.y = cval ? b1 : 0.0f;
            acc = __builtin_amdgcn_wmma_f32_16x16x4_f32(
                false, a, false, b, (short)0, acc, false, false);
        }
    }

    // out[bag] : row (a,b) = v + 8*hi, col c = lrow (<4) -> d = a*16 + b*4 + c
    float* obag = out + (size_t)bag * 64;
    if (cval) {
#pragma unroll
        for (int v = 0; v < 8; ++v)
            obag[(v + 8 * hi) * 4 + lrow] = acc[v];
    }
}

// ---------------------------------------------------------------------------
extern "C" void kernel_launch(void* const* d_in, const int* in_sizes, int n_in,
                              void* d_out, int out_size, void* d_ws, size_t ws_size,
                              hipStream_t stream) {
    (void)in_sizes; (void)n_in; (void)out_size; (void)ws_size;
    const int*   indices = (const int*)d_in[0];
    const int*   offsets = (const int*)d_in[1];
    const float* core0   = (const float*)d_in[2];
    const float* core1   = (const float*)d_in[3];
    const float* core2   = (const float*)d_in[4];
    float* out = (float*)d_out;
    float* w   = (float*)d_ws;   // requires 4*25*100*2048*4 = 81,920,000 B of scratch

    // Phase 1: 10,000 dense pair-contractions (fully utilized WMMA tiles)
    tt_phase1_kernel<<<NTAB * I0G * NP, 256, 0, stream>>>(core0, core1, w);
    // Phase 2: 32,768 bags, 8 bags (waves) per block, pooling fused into K
    tt_phase2_kernel<<<(NTAB * 8192) / 8, 256, 0, stream>>>(indices, offsets, core2, w, out);
}